// Net0_1778116460641
// MI455X (gfx1250) — compile-verified
//
#include <hip/hip_runtime.h>

typedef float v2f __attribute__((ext_vector_type(2)));
typedef float v4f __attribute__((ext_vector_type(4)));
typedef float v8f __attribute__((ext_vector_type(8)));

#define NMID 19   // fc2..fc20
#define HID  7

// ---------------------------------------------------------------------------
// Prep: pack Wm[19][7][7] into the V_WMMA_F32_16X16X4_F32 A-operand image.
// Two K-chunks per layer (K=0..3 and K=4..7), one b128 (4 dwords) per lane:
//   lane j   (j<16): { W[j][0], W[j][1], W[j][4], W[j][5] }   (rows j>=7 -> 0)
//   lane j+16:       { W[j][2], W[j][3], W[j][6], 0 }         (K=7 padded 0)
// 128 floats per layer -> 19*128 floats in d_ws.
// ---------------------------------------------------------------------------
__global__ void prep_pack(const float* __restrict__ Wm, float* __restrict__ wpack) {
    int t = blockIdx.x * blockDim.x + threadIdx.x;       // [19][32][4]
    if (t >= NMID * 32 * 4) return;
    int l    = t >> 7;
    int lane = (t >> 2) & 31;
    int c    = t & 3;
    int row  = lane & 15;
    int k    = (lane < 16) ? ((c < 2) ? c : c + 2)                  // 0,1,4,5
                           : ((c == 0) ? 2 : (c == 1) ? 3
                                           : (c == 2) ? 6 : 7);     // 2,3,6,pad
    float v = 0.0f;
    if (row < HID && k < HID) v = Wm[l * 49 + row * 7 + k];
    wpack[t] = v;
}

// ---------------------------------------------------------------------------
// Main fused MLP kernel: one wave = one 16-row tile.
// Batch in N, features in M; two f32 WMMAs per layer (K=0..3, K=4..7).
// ---------------------------------------------------------------------------
__device__ __forceinline__ float elu_f(float t) {
    float e = __expf(t) - 1.0f;          // v_mul + v_exp_f32 + v_add
    return t > 0.0f ? t : e;             // v_cmp + v_cndmask
}

__device__ __forceinline__ float swap16(float x) {
    // SWAPX16: lane L reads lane L^16 (and=0x1f, or=0, xor=0x10 -> 0x401F)
    return __int_as_float(__builtin_amdgcn_ds_swizzle(__float_as_int(x), 0x401F));
}

__global__ __launch_bounds__(256) void mlp_kernel(
    const float* __restrict__ x,  const float* __restrict__ W1,
    const float* __restrict__ b1, const float* __restrict__ bm,
    const float* __restrict__ Wo, const float* __restrict__ bo,
    const float* __restrict__ wpack, float* __restrict__ out, int nrows)
{
    const int gid  = blockIdx.x * blockDim.x + threadIdx.x;
    const int wave = gid >> 5;
    const int lane = threadIdx.x & 31;
    const int col  = lane & 15;
    const int row0 = wave * 16;
    if (row0 >= nrows) return;                   // wave-uniform exit
    int row = row0 + col;
    if (row >= nrows) row = nrows - 1;           // clamp (nrows % 16 == 0 normally)

    // ---- fc1 + ELU (f32 VALU, K=2): each lane computes its column's 7 features
    v2f xv = *(const v2f*)(x + (size_t)row * 2);
    float m[7];
#pragma unroll
    for (int j = 0; j < 7; ++j) {
        float t = fmaf(xv.y, W1[2 * j + 1], fmaf(xv.x, W1[2 * j], b1[j]));
        m[j] = elu_f(t);
    }

    const bool lo16 = lane < 16;
    v8f dzero = {};                               // zero C operand
    v4f wnext = *(const v4f*)(wpack + (size_t)lane * 4);   // layer 0 weights

    // ---- fc2..fc20: WMMA chain, batch in N, features in M
#pragma unroll 1
    for (int l = 0; l < NMID; ++l) {
        v4f w = wnext;
        int ln = (l < NMID - 1) ? (l + 1) : l;    // prefetch next layer's A image
        wnext = *(const v4f*)(wpack + (size_t)ln * 128 + (size_t)lane * 4);

        // Build B operand (4x16 f32 per chunk): lanes 0-15 hold K even-base rows,
        // lanes 16-31 hold the mirrored K rows brought over with ds_swizzle.
        float s2 = swap16(m[2]);
        float s3 = swap16(m[3]);
        float s6 = swap16(m[6]);
        v2f B0 = { lo16 ? m[0] : s2,  lo16 ? m[1] : s3   };  // K=0..3
        v2f B1 = { lo16 ? m[4] : s6,  lo16 ? m[5] : 0.0f };  // K=4..7 (K=7 pad)
        v2f A0 = { w.x, w.y };                                // W chunk K=0..3
        v2f A1 = { w.z, w.w };                                // W chunk K=4..7

        v8f d = __builtin_amdgcn_wmma_f32_16x16x4_f32(
                    false, A0, false, B0, (short)0, dzero, false, false);
        d     = __builtin_amdgcn_wmma_f32_16x16x4_f32(
                    false, A1, false, B1, (short)0, d,     false, false);

        // bias + ELU (lane-local: VGPR r of D == feature r of this lane's column)
        const float* bl = bm + l * 7;             // uniform -> scalar loads
#pragma unroll
        for (int r = 0; r < 7; ++r)
            m[r] = elu_f(d[r] + bl[r]);
    }

    // ---- fc21 + log_softmax (f32 VALU, N=2)
    float l0 = bo[0], l1 = bo[1];
#pragma unroll
    for (int k = 0; k < 7; ++k) {
        l0 = fmaf(m[k], Wo[k],     l0);
        l1 = fmaf(m[k], Wo[7 + k], l1);
    }
    float mx = fmaxf(l0, l1);
    float a0 = l0 - mx, a1 = l1 - mx;
    float ls = __logf(__expf(a0) + __expf(a1));
    if (lo16 && (row0 + col) < nrows) {
        v2f o = { a0 - ls, a1 - ls };
        *(v2f*)(out + (size_t)row * 2) = o;
    }
}

// ---------------------------------------------------------------------------
extern "C" void kernel_launch(void* const* d_in, const int* in_sizes, int n_in,
                              void* d_out, int out_size, void* d_ws, size_t ws_size,
                              hipStream_t stream) {
    const float* x  = (const float*)d_in[0];
    const float* W1 = (const float*)d_in[1];
    const float* b1 = (const float*)d_in[2];
    const float* Wm = (const float*)d_in[3];
    const float* bm = (const float*)d_in[4];
    const float* Wo = (const float*)d_in[5];
    const float* bo = (const float*)d_in[6];
    float* out   = (float*)d_out;
    float* wpack = (float*)d_ws;                 // 19*128 floats = 9728 B

    const int nrows = in_sizes[0] / 2;

    prep_pack<<<(NMID * 128 + 255) / 256, 256, 0, stream>>>(Wm, wpack);

    const int tiles  = (nrows + 15) / 16;        // one wave per 16-row tile
    const int blocks = (tiles + 7) / 8;          // 8 waves per 256-thread block
    mlp_kernel<<<blocks, 256, 0, stream>>>(x, W1, b1, bm, Wo, bo, wpack, out, nrows);
}